// Net_81862076662284
// MI455X (gfx1250) — compile-verified
//
#include <hip/hip_runtime.h>

typedef float v2f __attribute__((ext_vector_type(2)));
typedef float v4f __attribute__((ext_vector_type(4)));
typedef float v8f __attribute__((ext_vector_type(8)));
typedef int   v4i_gcc __attribute__((vector_size(16)));   // matches builtin param

#define CIN   16
#define COUT  32
#define DOUT  126
#define NOFF  27
#define WELEMS (NOFF * CIN * COUT)   // 13824 floats = 54 KB

// One wave32 handles a tile of 16 voxels. For each of the 27 kernel offsets it
// computes the 16x16(xCIN=16) x (16x32) GEMM with v_wmma_f32_16x16x4_f32
// (4 K-steps x 2 N-tiles = 8 WMMAs), then scatter-adds its C fragment into the
// dense output with fp32 global atomics, masking invalid (out-of-grid) rows.
__global__ __launch_bounds__(256)
void spconv_wmma_kernel(const float* __restrict__ features,
                        const int*   __restrict__ coors,
                        const float* __restrict__ weight,
                        float*       __restrict__ out,
                        unsigned char* __restrict__ mask,
                        int nvox, int ntiles)
{
    __shared__ float wlds[WELEMS];
    const int tid = threadIdx.x;

    // ---- Stage all 27 [16x32] weight slices into LDS (54 KB of 320 KB/WGP) ----
#if __has_builtin(__builtin_amdgcn_global_load_async_to_lds_b128) && \
    __has_builtin(__builtin_amdgcn_s_wait_asynccnt)
    // CDNA5 async copy: memory -> LDS without passing through VGPRs (ASYNCcnt).
    {
        const int nvec = WELEMS / 4;               // 3456 x b128 transfers
        for (int i = tid; i < nvec; i += 256) {
            __builtin_amdgcn_global_load_async_to_lds_b128(
                (__attribute__((address_space(1))) v4i_gcc*)(weight + 4 * i),
                (__attribute__((address_space(3))) v4i_gcc*)(wlds + 4 * i),
                0 /*imm offset*/, 0 /*cpol*/);
        }
        __builtin_amdgcn_s_wait_asynccnt(0);
    }
#else
    {
        const int nvec = WELEMS / 4;
        for (int i = tid; i < nvec; i += 256)
            *(v4f*)(wlds + 4 * i) = *(const v4f*)(weight + 4 * i);
    }
#endif
    __syncthreads();

    const int wave = tid >> 5;
    const int lane = tid & 31;
    const int tile = blockIdx.x * 8 + wave;
    if (tile >= ntiles) return;           // whole-wave uniform exit

    const int v0   = tile * 16;
    const int half = lane >> 4;           // 0: lanes 0-15, 1: lanes 16-31
    const int l15  = lane & 15;

    // ---- A fragments: 16x16 fp32 feature tile, ISA 16x4 layout per step ----
    // lane row = lane&15 ; VGPR0/1 hold K = 4*kk + 2*half, +1
    v2f a[4];
    {
        const int arow = v0 + l15;
        if (arow < nvox) {
            const float* rp = features + (long long)arow * CIN + half * 2;
            #pragma unroll
            for (int kk = 0; kk < 4; ++kk)
                a[kk] = *(const v2f*)(rp + 4 * kk);
        } else {
            #pragma unroll
            for (int kk = 0; kk < 4; ++kk) a[kk] = (v2f){0.f, 0.f};
        }
    }

    // ---- coordinates for the 8 C rows this lane owns (row = r + 8*half) ----
    int cz[8], cy[8], cx[8];
    #pragma unroll
    for (int r = 0; r < 8; ++r) {
        const int crow = v0 + r + half * 8;
        if (crow < nvox) {
            const int* cp = coors + (long long)crow * 4;
            cz[r] = cp[1]; cy[r] = cp[2]; cx[r] = cp[3];
        } else {
            cz[r] = -100000; cy[r] = -100000; cx[r] = -100000;
        }
    }

    const int col  = l15;        // C/D and B column for this lane
    const int krow = half * 2;   // B K-row pair base for this lane

    for (int k = 0; k < NOFF; ++k) {
        const int kz = k / 9, ky = (k / 3) % 3, kx = k % 3;
        const float* wk = wlds + k * (CIN * COUT);

        v8f c0 = {};  // output columns 0-15
        v8f c1 = {};  // output columns 16-31
        #pragma unroll
        for (int kk = 0; kk < 4; ++kk) {
            const int kb = 4 * kk + krow;
            v2f b0, b1;
            b0.x = wk[(kb    ) * COUT + col];
            b0.y = wk[(kb + 1) * COUT + col];
            b1.x = wk[(kb    ) * COUT + col + 16];
            b1.y = wk[(kb + 1) * COUT + col + 16];
            c0 = __builtin_amdgcn_wmma_f32_16x16x4_f32(false, a[kk], false, b0,
                                                       (short)0, c0, false, false);
            c1 = __builtin_amdgcn_wmma_f32_16x16x4_f32(false, a[kk], false, b1,
                                                       (short)0, c1, false, false);
        }

        // ---- scatter-add: lane owns rows r+8*half, one per C VGPR ----
        #pragma unroll
        for (int r = 0; r < 8; ++r) {
            const int oz = cz[r] - kz;
            const int oy = cy[r] - ky;
            const int ox = cx[r] - kx;
            if ((unsigned)oz < DOUT && (unsigned)oy < DOUT && (unsigned)ox < DOUT) {
                const long long site = ((long long)oz * DOUT + oy) * DOUT + ox;
                float* op = out + site * COUT;
                atomicAdd(op + col,      c0[r]);
                atomicAdd(op + col + 16, c1[r]);
                if (col == 0) mask[site] = 1;   // benign race: all write 1
            }
        }
    }
}

// bias only at active output sites; inactive sites stay exactly zero
__global__ __launch_bounds__(256)
void bias_kernel(float* __restrict__ out,
                 const unsigned char* __restrict__ mask,
                 const float* __restrict__ bias,
                 long long n)
{
    const long long idx = (long long)blockIdx.x * 256 + threadIdx.x;
    if (idx >= n) return;
    if (mask[idx >> 5]) out[idx] += bias[idx & 31];
}

extern "C" void kernel_launch(void* const* d_in, const int* in_sizes, int n_in,
                              void* d_out, int out_size, void* d_ws, size_t ws_size,
                              hipStream_t stream)
{
    const float* features = (const float*)d_in[0];
    const int*   coors    = (const int*)  d_in[1];
    const float* weight   = (const float*)d_in[2];
    const float* bias     = (const float*)d_in[3];
    // d_in[4] = batch_size (1), unused

    float*         out  = (float*)d_out;
    unsigned char* mask = (unsigned char*)d_ws;

    const int nvox        = in_sizes[0] / CIN;            // 200000
    const long long nout  = (long long)out_size;          // 126^3 * 32
    const long long nsite = nout / COUT;                  // 126^3

    // zero dense output + active mask (capturable async memsets)
    (void)hipMemsetAsync(out,  0, (size_t)nout * sizeof(float), stream);
    (void)hipMemsetAsync(mask, 0, (size_t)nsite, stream);

    const int ntiles  = (nvox + 15) / 16;
    const int nblocks = (ntiles + 7) / 8;                 // 8 waves / block
    spconv_wmma_kernel<<<nblocks, 256, 0, stream>>>(features, coors, weight,
                                                    out, mask, nvox, ntiles);

    const int bblocks = (int)((nout + 255) / 256);
    bias_kernel<<<bblocks, 256, 0, stream>>>(out, mask, bias, nout);
}